// GCN_5944234737723
// MI455X (gfx1250) — compile-verified
//
#include <hip/hip_runtime.h>

typedef float v2f __attribute__((ext_vector_type(2)));
typedef float v8f __attribute__((ext_vector_type(8)));

#define DFEAT 128

// ---------------------------------------------------------------------------
// Phase 0: zero the accumulator (d_out doubles as h)
// ---------------------------------------------------------------------------
__global__ void gcn_zero_kernel(float4* __restrict__ p, int n4) {
    int i = blockIdx.x * blockDim.x + threadIdx.x;
    int stride = gridDim.x * blockDim.x;
    for (; i < n4; i += stride)
        p[i] = make_float4(0.f, 0.f, 0.f, 0.f);
}

// ---------------------------------------------------------------------------
// Phase 1: edge gather + scatter-add. One wave32 per edge:
//   32 lanes x float4 = one 128-float feature row.
//   Gather is a fully coalesced 512B global_load_b128 per wave (L2-resident),
//   scatter is 4 global_atomic_add_f32 per lane (no-return -> STOREcnt path).
// ---------------------------------------------------------------------------
__global__ void __launch_bounds__(256)
gcn_scatter_kernel(const float* __restrict__ feature,
                   const int* __restrict__ src,
                   const int* __restrict__ dst,
                   float* __restrict__ h,
                   int n_edges) {
    int e = blockIdx.x * 8 + (threadIdx.x >> 5);   // 8 waves per block
    int lane = threadIdx.x & 31;
    if (e >= n_edges) return;
    int s = src[e];
    int d = dst[e];
    const float4* frow = (const float4*)(feature + (size_t)s * DFEAT);
    float4 v = frow[lane];
    float* hrow = h + (size_t)d * DFEAT + lane * 4;
    atomicAdd(hrow + 0, v.x);
    atomicAdd(hrow + 1, v.y);
    atomicAdd(hrow + 2, v.z);
    atomicAdd(hrow + 3, v.w);
}

// ---------------------------------------------------------------------------
// Phase 2: out = relu(h @ W^T + b), in place on h (=d_out).
// One wave32 per 16-node tile; 8 accumulators cover all 128 outputs.
// fp32 WMMA 16x16x4; K=128 stepped by 4.
//
// A-fragment layout (32-bit A 16x4, ISA 7.12.2):
//   lane<16 : a = { h[node0+lane][k+0], h[node0+lane][k+1] }
//   lane>=16: a = { h[node0+lane-16][k+2], h[node0+lane-16][k+3] }
// B-fragment (B = W^T, i.e. B[k][o] = W[o][k]), mirrored layout:
//   lane<16 : b = { B[k+0][o], B[k+1][o] },  o = tile*16 + lane
//   lane>=16: b = { B[k+2][o], B[k+3][o] }
// W^T staged in LDS as Wt[k][o] so the 16-lane fragment reads are
// consecutive dwords -> bank-conflict-free ds_load_b32.
// C/D layout: acc[t][v] -> row node0 + v + 8*(lane>>4), col t*16 + (lane&15).
// ---------------------------------------------------------------------------
__global__ void __launch_bounds__(128)
gcn_gemm_relu_kernel(float* __restrict__ hio,
                     const float* __restrict__ W,
                     const float* __restrict__ bias,
                     int n_nodes) {
    __shared__ float Wt[DFEAT * DFEAT];   // 64 KB: Wt[k*128 + o] = W[o*128 + k]

    // Cooperative transposed load of W (LDS writes are coalesced/conflict-free).
    for (int idx = threadIdx.x; idx < DFEAT * DFEAT; idx += 128) {
        int k = idx >> 7;
        int o = idx & 127;
        Wt[idx] = W[o * DFEAT + k];
    }
    __syncthreads();

    int wave  = threadIdx.x >> 5;
    int lane  = threadIdx.x & 31;
    int tile  = blockIdx.x * 4 + wave;
    int node0 = tile * 16;
    if (node0 >= n_nodes) return;

    int lhalf = lane >> 4;   // 0 | 1
    int lmod  = lane & 15;

    v8f acc[8];
    v8f zero = {};
#pragma unroll
    for (int t = 0; t < 8; ++t) acc[t] = zero;

    const float* arow = hio + (size_t)(node0 + lmod) * DFEAT + 2 * lhalf;

#pragma unroll 1
    for (int k = 0; k < DFEAT; k += 4) {
        v2f a;
        a.x = arow[k + 0];
        a.y = arow[k + 1];
        int kb = (k + 2 * lhalf) * DFEAT;
#pragma unroll
        for (int t = 0; t < 8; ++t) {
            int o = t * 16 + lmod;
            v2f b;
            b.x = Wt[kb + o];
            b.y = Wt[kb + DFEAT + o];
            acc[t] = __builtin_amdgcn_wmma_f32_16x16x4_f32(
                false, a, false, b, (short)0, acc[t], false, false);
        }
    }

    // bias + relu + in-place store (this wave owns rows node0..node0+15;
    // all its loads precede these stores in program order)
#pragma unroll
    for (int t = 0; t < 8; ++t) {
        int col = t * 16 + lmod;
        float bv = bias[col];
#pragma unroll
        for (int v = 0; v < 8; ++v) {
            int row = node0 + v + 8 * lhalf;
            float val = acc[t][v] + bv;
            hio[(size_t)row * DFEAT + col] = val > 0.f ? val : 0.f;
        }
    }
}

// ---------------------------------------------------------------------------
extern "C" void kernel_launch(void* const* d_in, const int* in_sizes, int n_in,
                              void* d_out, int out_size, void* d_ws, size_t ws_size,
                              hipStream_t stream) {
    const float* feature = (const float*)d_in[0];
    const int*   src     = (const int*)d_in[1];
    const int*   dst     = (const int*)d_in[2];
    const float* W       = (const float*)d_in[3];
    const float* bias    = (const float*)d_in[4];
    float*       out     = (float*)d_out;

    int n_nodes = in_sizes[0] / DFEAT;   // 100000
    int n_edges = in_sizes[1];           // 3200000

    // Phase 0: zero h (= d_out)
    int n4 = n_nodes * (DFEAT / 4);
    gcn_zero_kernel<<<2048, 256, 0, stream>>>((float4*)out, n4);

    // Phase 1: scatter-add, 8 edges (waves) per 256-thread block
    int eblocks = (n_edges + 7) / 8;
    gcn_scatter_kernel<<<eblocks, 256, 0, stream>>>(feature, src, dst, out, n_edges);

    // Phase 2: WMMA GEMM + bias + relu, in place on d_out
    int tiles   = (n_nodes + 15) / 16;   // 6250
    int gblocks = (tiles + 3) / 4;
    gcn_gemm_relu_kernel<<<gblocks, 128, 0, stream>>>(out, W, bias, n_nodes);
}